// SplineConv_7146825580908
// MI455X (gfx1250) — compile-verified
//
#include <hip/hip_runtime.h>

// SplineConv as dense matmul: out (N x 1024) = S (N x 64) x CtrlT (64 x 1024)
// where S[n] is the sparse tensor-product B-spline basis row (9 nonzeros).
// fp32 WMMA (V_WMMA_F32_16X16X4_F32), K looped 64/4 = 16 chunks.
// CDNA5 paths: v_wmma f32, GLOBAL_LOAD_ASYNC_TO_LDS_B128 staging (ASYNCcnt),
// non-temporal streaming stores for the write-once 64MB output.

typedef __attribute__((ext_vector_type(2))) float v2f;
typedef __attribute__((ext_vector_type(4))) float v4f;
typedef __attribute__((ext_vector_type(8))) float v8f;

#define NCH        1024   // out_ch * in_group
#define KP         64     // 8*8 control grid flattened
#define PTS_PER_WG 128    // 8 waves x 16-point tiles
#define CH_PER_WG  128    // 8 channel tiles of 16
#define LDSB_STRIDE 72    // padded row stride (floats) for Ctrl rows in LDS

#if __has_builtin(__builtin_amdgcn_global_load_async_to_lds_b128) && \
    __has_builtin(__builtin_amdgcn_s_wait_asynccnt)
#define HAVE_ASYNC_LDS 1
// Param types per hipcc diagnostic: addrspace(1) ptr to vector_size(16) int.
typedef int async_v4i __attribute__((vector_size(16)));
typedef __attribute__((address_space(1))) async_v4i g_v4i;
typedef __attribute__((address_space(3))) async_v4i l_v4i;
#else
#define HAVE_ASYNC_LDS 0
#endif

// de Boor order-2 basis weights (matches reference recursion, den==0 -> 1)
__device__ inline void basis3(const float* __restrict__ T, int k, float x,
                              float& B0, float& B1, float& B2) {
    float d10 = T[k + 1] - T[k - 1];
    float d11 = T[k + 2] - T[k];
    float d21 = T[k + 1] - T[k];
    float a10 = (x - T[k - 1]) / ((d10 == 0.0f) ? 1.0f : d10);
    float a11 = (x - T[k])     / ((d11 == 0.0f) ? 1.0f : d11);
    float a21 = (x - T[k])     / ((d21 == 0.0f) ? 1.0f : d21);
    B0 = (1.0f - a21) * (1.0f - a10);
    B1 = (1.0f - a21) * a10 + a21 * (1.0f - a11);
    B2 = a21 * a11;
}

__global__ __launch_bounds__(256) void spline_wmma_kernel(
    const float* __restrict__ xy,   // (N,2)
    const float* __restrict__ C,    // (1024,64) = (oc,ig,h,w)
    const float* __restrict__ Tx,   // (11)
    const float* __restrict__ Ty,   // (11)
    float* __restrict__ out,        // (N,1024)
    int npts)
{
    __shared__ float ldsB[CH_PER_WG * LDSB_STRIDE]; // [c_local][p], padded
    __shared__ float pdat[PTS_PER_WG][8];           // Bx0..2,By0..2,kx,ky

    const int tid = threadIdx.x;
    const int n0  = blockIdx.x * PTS_PER_WG; // point base
    const int c0  = blockIdx.y * CH_PER_WG;  // channel base

    // ---- Stage Ctrl rows [c0, c0+128) into LDS (channel-major, padded) ----
    #pragma unroll
    for (int i = 0; i < 8; ++i) {
        int idx = tid + i * 256;        // 0..2047 float4 slots
        int row = idx >> 4;             // 16 float4 per 64-float row
        int col = (idx & 15) << 2;
        const float* g = C + (size_t)(c0 + row) * KP + col;
        float*       l = &ldsB[row * LDSB_STRIDE + col];
#if HAVE_ASYNC_LDS
        // Direct memory->LDS DMA, tracked with ASYNCcnt (no VGPR round trip).
        __builtin_amdgcn_global_load_async_to_lds_b128(
            (g_v4i*)(unsigned long long)g,
            (l_v4i*)(unsigned int)(unsigned long long)l,
            /*offset=*/0, /*cpol=*/0);
#else
        *(v4f*)l = *(const v4f*)g;
#endif
    }

    // ---- Per-point basis weights (threads 0..127, one point each) ----
    if (tid < PTS_PER_WG) {
        int n = n0 + tid;
        float x = xy[2 * n + 0];
        float y = xy[2 * n + 1];
        int cx = 0, cy = 0;
        #pragma unroll
        for (int t = 0; t < 11; ++t) {
            cx += (Tx[t] <= x) ? 1 : 0;
            cy += (Ty[t] <= y) ? 1 : 0;
        }
        int kx = min(max(cx - 1, 2), 7);   // clip(searchsorted-1, px, len-px-2)
        int ky = min(max(cy - 1, 2), 7);
        float bx0, bx1, bx2, by0, by1, by2;
        basis3(Tx, kx, x, bx0, bx1, bx2);
        basis3(Ty, ky, y, by0, by1, by2);
        pdat[tid][0] = bx0; pdat[tid][1] = bx1; pdat[tid][2] = bx2;
        pdat[tid][3] = by0; pdat[tid][4] = by1; pdat[tid][5] = by2;
        pdat[tid][6] = __int_as_float(kx);
        pdat[tid][7] = __int_as_float(ky);
    }

#if HAVE_ASYNC_LDS
    __builtin_amdgcn_s_wait_asynccnt(0);   // async LDS writes complete
#endif
    __syncthreads();

    // ---- WMMA: each wave handles a 16-point tile x 128 channels ----
    const int lane = tid & 31;
    const int wave = tid >> 5;        // 0..7 -> point tile within block
    const int m    = lane & 15;       // M (for A) / N (for B,D) index
    const int hi   = lane >> 4;       // half-wave selector

    // Basis data for this lane's point (lanes L and L+16 share a point)
    const float* pd = pdat[wave * 16 + m];
    const float bx0 = pd[0], bx1 = pd[1], bx2 = pd[2];
    const float by0 = pd[3], by1 = pd[4], by2 = pd[5];
    const int   kx  = __float_as_int(pd[6]);
    const int   ky  = __float_as_int(pd[7]);
    const int   hbase = kx - 2, wbase = ky - 2;

    // A fragments: 16 K-chunks. ISA layout (32-bit A 16x4):
    // lane L: M = L%16; vgpr e holds K = 4*kc + 2*(L/16) + e.
    v2f afrag[16];
    #pragma unroll
    for (int kc = 0; kc < 16; ++kc) {
        #pragma unroll
        for (int e = 0; e < 2; ++e) {
            int k  = kc * 4 + hi * 2 + e;       // p = h*8 + w in 0..63
            int h  = k >> 3;
            int w  = k & 7;
            int dh = h - hbase;
            int dw = w - wbase;
            float bx = (dh == 0) ? bx0 : (dh == 1) ? bx1 : (dh == 2) ? bx2 : 0.0f;
            float by = (dw == 0) ? by0 : (dw == 1) ? by1 : (dw == 2) ? by2 : 0.0f;
            afrag[kc][e] = bx * by;
        }
    }

    // B fragment base: row = channel c_local, cols = K pair (contiguous b64)
    const float* bbase = &ldsB[0] + hi * 2;

    for (int ct = 0; ct < 8; ++ct) {
        const float* brow = bbase + (ct * 16 + m) * LDSB_STRIDE;
        v8f acc = {};
        #pragma unroll
        for (int kc = 0; kc < 16; ++kc) {
            v2f b = *(const v2f*)(brow + kc * 4);
            acc = __builtin_amdgcn_wmma_f32_16x16x4_f32(
                /*neg_a=*/false, afrag[kc],
                /*neg_b=*/false, b,
                /*c_mod=*/(short)0, acc,
                /*reuse_a=*/false, /*reuse_b=*/false);
        }
        // D layout: vgpr v -> M = v + 8*hi, N = m. Output is a write-once
        // 64MB stream -> non-temporal stores (TH=NT), keep L2 for Ctrl.
        const int c     = c0 + ct * 16 + m;
        const int nrow0 = n0 + wave * 16 + hi * 8;
        #pragma unroll
        for (int v = 0; v < 8; ++v) {
            __builtin_nontemporal_store(acc[v],
                out + (size_t)(nrow0 + v) * NCH + c);
        }
    }
}

extern "C" void kernel_launch(void* const* d_in, const int* in_sizes, int n_in,
                              void* d_out, int out_size, void* d_ws, size_t ws_size,
                              hipStream_t stream) {
    const float* xy = (const float*)d_in[0];
    const float* C  = (const float*)d_in[1];
    const float* Tx = (const float*)d_in[2];
    const float* Ty = (const float*)d_in[3];
    float* out = (float*)d_out;

    int npts = in_sizes[0] / 2;                 // 16384
    dim3 grid((npts + PTS_PER_WG - 1) / PTS_PER_WG, NCH / CH_PER_WG);
    spline_wmma_kernel<<<grid, 256, 0, stream>>>(xy, C, Tx, Ty, out, npts);
}